// ExponentialMovingAverage_31516470018371
// MI455X (gfx1250) — compile-verified
//
#include <hip/hip_runtime.h>

// EMA scan  y[t] = w*x[t] + (1-w)*y[t-1],  y[0]=x[0]
// B=32, T=32768, C=40, fp32. Bandwidth-bound (336 MB min traffic ~14.4us @ 23.3TB/s).
// Strategy: scan-as-matmul on V_WMMA_F32_16X16X4_F32 (constant per-channel decay
// makes a 16-step segment a 16x16 lower-triangular matmul), 3-phase chunked scan
// so the input is read from HBM once (phase-3 re-read is L2-resident: 168MB < 192MB L2).

typedef float v2f __attribute__((ext_vector_type(2)));
typedef float v8f __attribute__((ext_vector_type(8)));

#define B_   32
#define T_   32768
#define C_   40
#define W_   256                 // timesteps per window (one WMMA tile per channel)
#define G_   (T_ / W_)           // 128 windows per sequence
#define NSEQ (B_ * C_)           // 1280 independent sequences
#define LDSF 10304               // 256*40 floats + swizzle headroom

__device__ __forceinline__ float clamp01(float w) {
  return fminf(fmaxf(w, 0.f), 1.f);
}

// MODE 0: compute per-window local finals F (carry-in = 0), no output write.
// MODE 1: read carry-in per window, write full scan output.
template <int MODE>
__global__ __launch_bounds__(256) void ema_window(
    const float* __restrict__ x, const float* __restrict__ wts,
    const float* __restrict__ cin, float* __restrict__ out,
    float* __restrict__ Fout)
{
  __shared__ float lds[LDSF];
  const int blk = blockIdx.x;
  const int b   = blk / G_;
  const int g   = blk % G_;
  const size_t base = ((size_t)b * T_ + (size_t)g * W_) * C_;  // floats
  const int tid = threadIdx.x;

  // Stage-in: 256*40 floats, contiguous in global -> LDS with a +4*(t>>4) dword
  // swizzle so the chunk-strided (640-dword) WMMA B-operand reads spread over
  // 16 LDS banks instead of all hitting one bank.
  for (int i = tid; i < (W_ * C_) / 4; i += 256) {
    float4 v = reinterpret_cast<const float4*>(x + base)[i];
    int flat = i * 4;
    int t    = flat / C_;                  // c = flat % 40 is a multiple of 4
    int sw   = flat + ((t >> 4) << 2);     // stays 16B aligned
    *reinterpret_cast<float4*>(&lds[sw]) = v;
  }
  __syncthreads();

  const int wave = tid >> 5;
  const int lane = tid & 31;
  const int n    = lane & 15;   // WMMA column = chunk index (0..15) / A-row index
  const int h    = lane >> 4;   // half-wave

  for (int ci = 0; ci < 5; ++ci) {
    const int   c  = wave * 5 + ci;                 // 8 waves * 5 channels = 40
    const float w  = clamp01(wts[c]);
    const float a  = 1.f - w;
    const float a2 = a * a, a4 = a2 * a2, a8 = a4 * a4, a16 = a8 * a8;
    auto apow = [&](int e) {                        // a^e, e in [0,16]
      float r = 1.f;
      if (e & 1)  r *= a;
      if (e & 2)  r *= a2;
      if (e & 4)  r *= a4;
      if (e & 8)  r *= a8;
      if (e & 16) r *= a16;
      return r;
    };

    // A (16x4 slices of M_c, lane m=n holds row m; vgpr v + half h -> col v+2h):
    //   M[i][j] = (j<=i) ? w*a^(i-j) : 0
    // B (4x16 slices of X, row K = v+2h in vgpr v/half h, col = chunk n):
    //   X[j][n] = x[t0 + 16n + j]  (read from swizzled LDS; t>>4 == n here)
    v2f A[4], Bv[4];
#pragma unroll
    for (int k = 0; k < 4; ++k) {
#pragma unroll
      for (int v = 0; v < 2; ++v) {
        int j   = 4 * k + v + 2 * h;
        A[k][v] = (j <= n) ? w * apow(n - j) : 0.f;
        int t   = 16 * n + 4 * k + v + 2 * h;
        Bv[k][v] = lds[t * C_ + c + 4 * n];
      }
    }

    // Ylocal(16x16) = M_c * X via 4 chained 16x16x4 f32 WMMAs (D->C accumulate).
    v8f acc = {};
#pragma unroll
    for (int k = 0; k < 4; ++k)
      acc = __builtin_amdgcn_wmma_f32_16x16x4_f32(
          false, A[k], false, Bv[k], (short)0, acc, false, false);

    // Chunk-local final = row 15 of D (VGPR 7, high half, col = lane-16).
    float Fn = __shfl(acc[7], 16 + n, 32);

    // Compose the 16 chunk operators x -> a16*x + Fn with a log2(16) shuffle
    // scan over (decay d, offset s) pairs: R∘E = (dR*dE, dR*sE + sR).
    float d = a16, s = Fn;
#pragma unroll
    for (int o = 1; o < 16; o <<= 1) {
      float dn = __shfl(d, lane - o, 32);
      float sn = __shfl(s, lane - o, 32);
      if (n >= o) { s = s + d * sn; d = d * dn; }
    }

    if (MODE == 0) {
      // Window-local final (carry-in 0) = inclusive offset at chunk 15.
      if (lane == 15) Fout[(size_t)g * NSEQ + b * C_ + c] = s;
    } else {
      const float Cin = cin[(size_t)g * NSEQ + b * C_ + c];
      float sp = __shfl(s, lane - 1, 32);
      float dp = __shfl(d, lane - 1, 32);
      float P  = (n == 0) ? Cin : (sp + dp * Cin);   // carry into chunk n
      // Y[i][n] = Ylocal[i][n] + a^(i+1) * P ; write back through LDS (this
      // wave owns channel c's slots, which it has finished reading).
#pragma unroll
      for (int r = 0; r < 8; ++r) {
        int   i = r + 8 * h;                          // D row for this vgpr/half
        float y = acc[r] + apow(i + 1) * P;
        int   t = 16 * n + i;
        lds[t * C_ + c + 4 * n] = y;
      }
    }
  }

  if (MODE == 1) {
    __syncthreads();
    // Coalesced b128 store of the whole window (un-swizzle).
    for (int i = tid; i < (W_ * C_) / 4; i += 256) {
      int flat = i * 4;
      int t    = flat / C_;
      int sw   = flat + ((t >> 4) << 2);
      float4 v = *reinterpret_cast<float4*>(&lds[sw]);
      reinterpret_cast<float4*>(out + base)[i] = v;
    }
  }
}

// Phase 2: per-(b,c) lookback across the 128 windows; window operator is
// x -> a^256 * x + F_g. Virtual carry y[-1] = x[b,0,c] reproduces y[0]=x[0]
// exactly (a = 1-w, and for w==1 the carry term vanishes anyway).
__global__ __launch_bounds__(256) void ema_lookback(
    const float* __restrict__ x, const float* __restrict__ wts,
    const float* __restrict__ F, float* __restrict__ CIN)
{
  int bc = blockIdx.x * 256 + threadIdx.x;
  if (bc >= NSEQ) return;
  int b = bc / C_, c = bc % C_;
  float w  = clamp01(wts[c]);
  float a  = 1.f - w;
  float aW = a;
#pragma unroll
  for (int i = 0; i < 8; ++i) aW *= aW;              // a^256
  float carry = x[(size_t)b * T_ * C_ + c];          // virtual y[-1] = x[b,0,c]
  for (int g = 0; g < G_; ++g) {
    CIN[(size_t)g * NSEQ + bc] = carry;
    carry = F[(size_t)g * NSEQ + bc] + aW * carry;
  }
}

extern "C" void kernel_launch(void* const* d_in, const int* in_sizes, int n_in,
                              void* d_out, int out_size, void* d_ws, size_t ws_size,
                              hipStream_t stream) {
  const float* x   = (const float*)d_in[0];   // inputs [B,T,C]
  // d_in[1] = initial_state, unused by the reference math
  const float* wts = (const float*)d_in[2];   // weights [C]
  float* out = (float*)d_out;

  // Scratch: F (window-local finals) + CIN (per-window carry-in), 655KB each.
  float* F   = (float*)d_ws;
  float* CIN = F + (size_t)G_ * NSEQ;

  dim3 grid(B_ * G_), block(256);
  ema_window<0><<<grid, block, 0, stream>>>(x, wts, nullptr, nullptr, F);
  ema_lookback<<<(NSEQ + 255) / 256, 256, 0, stream>>>(x, wts, F, CIN);
  ema_window<1><<<grid, block, 0, stream>>>(x, wts, CIN, out, nullptr);
}